// DQSN_13726715478283
// MI455X (gfx1250) — compile-verified
//
#include <hip/hip_runtime.h>

typedef __attribute__((ext_vector_type(2))) float v2f;
typedef __attribute__((ext_vector_type(8))) float v8f;

#define TSTEPS 16
#define HDIM 4096
#define HTILES (HDIM / 16)               // 256 h-tiles
#define WAVES_PER_BLOCK 8
#define TILES_PER_WAVE (HTILES / WAVES_PER_BLOCK)  // 32

// One block = one 16-row batch tile; 8 waves split the H dimension.
// Per h-tile: WMMA computes the 16x16 I1 tile (K=4 f32 WMMA), then each lane
// simulates 16 IF steps for its column (8 batch rows in the C layout),
// producing weighted spike sums that are FMA'd against W2 into per-lane
// accumulators. Final: 16-lane butterfly + LDS atomics -> [16,2] output tile.
__global__ __launch_bounds__(256) void dqsn_fused_kernel(
    const float* __restrict__ x,   // [B,4]
    const float* __restrict__ W1,  // [H,4]
    const float* __restrict__ b1,  // [H]
    const float* __restrict__ W2,  // [2,H]
    const float* __restrict__ b2,  // [2]
    float* __restrict__ out)       // [B,2]
{
    __shared__ float red[32];      // [row*2 + ch] for the 16-row tile
    const int tid = threadIdx.x;
    if (tid < 32) red[tid] = 0.0f;
    __syncthreads();

    const int lane  = tid & 31;
    const int wave  = tid >> 5;
    const int lh    = lane & 15;   // column index within tile (N) / row M base
    const int half  = lane >> 4;   // lane half: selects K pair (A/B) and M+8 (C)
    const int bbase = blockIdx.x * 16;

    // A tile (x rows), ISA 16x4 f32 layout: lanes 0-15 hold K={0,1}, 16-31 K={2,3}
    v2f amat = *(const v2f*)(x + (bbase + lh) * 4 + half * 2);

    float acc0[8], acc1[8];
#pragma unroll
    for (int m = 0; m < 8; ++m) { acc0[m] = 0.0f; acc1[m] = 0.0f; }

    for (int i = 0; i < TILES_PER_WAVE; ++i) {
        const int hbase = (wave * TILES_PER_WAVE + i) * 16;
        const int h     = hbase + lh;          // this lane's column

        // B tile = W1^T (4x16): lane holds N=lh, K pair selected by half
        v2f bmat = *(const v2f*)(W1 + h * 4 + half * 2);

        v8f cz = {0.f, 0.f, 0.f, 0.f, 0.f, 0.f, 0.f, 0.f};
        // I1 tile: D = A(16x4) x B(4x16), f32 accumulate
        v8f c = __builtin_amdgcn_wmma_f32_16x16x4_f32(
            /*neg_a=*/false, amat, /*neg_b=*/false, bmat,
            /*c_mod=*/(short)0, cz, /*reuse_a=*/false, /*reuse_b=*/false);

        const float b1h = b1[h];
        float I[8], v[8], aval[8];
#pragma unroll
        for (int m = 0; m < 8; ++m) { I[m] = c[m] + b1h; v[m] = 0.0f; aval[m] = 0.0f; }

        // 16-step IF dynamics; aval accumulates 0.5^(17-t) per spike
        float w = 1.0f / 65536.0f;             // t=1 -> 2^-16
#pragma unroll
        for (int t = 0; t < TSTEPS; ++t) {
#pragma unroll
            for (int m = 0; m < 8; ++m) {
                v[m] += I[m];
                const bool s = v[m] >= 1.0f;   // V_TH = 1.0
                aval[m] += s ? w : 0.0f;
                v[m]    = s ? 0.0f : v[m];     // hard reset
            }
            w *= 2.0f;
        }

        const float w20 = W2[h];
        const float w21 = W2[HDIM + h];
#pragma unroll
        for (int m = 0; m < 8; ++m) {
            acc0[m] = fmaf(aval[m], w20, acc0[m]);
            acc1[m] = fmaf(aval[m], w21, acc1[m]);
        }
    }

    // Reduce over the 16 columns held by each lane-half (halves independent:
    // xor masks 1,2,4,8 never cross the 16-lane boundary).
#pragma unroll
    for (int m = 0; m < 8; ++m) {
#pragma unroll
        for (int off = 1; off < 16; off <<= 1) {
            acc0[m] += __shfl_xor(acc0[m], off, 32);
            acc1[m] += __shfl_xor(acc1[m], off, 32);
        }
    }

    // C layout: register m of half h holds batch row (h*8 + m).
#pragma unroll
    for (int m = 0; m < 8; ++m) {
        if (lh == m) {
            atomicAdd(&red[(half * 8 + m) * 2 + 0], acc0[m]);
            atomicAdd(&red[(half * 8 + m) * 2 + 1], acc1[m]);
        }
    }
    __syncthreads();

    if (tid < 32) {
        const float bias_scale = 1.0f - 1.0f / 65536.0f;   // sum_t 0.5^(17-t)
        out[bbase * 2 + tid] = red[tid] + b2[tid & 1] * bias_scale;
    }
}

extern "C" void kernel_launch(void* const* d_in, const int* in_sizes, int n_in,
                              void* d_out, int out_size, void* d_ws, size_t ws_size,
                              hipStream_t stream) {
    const float* x  = (const float*)d_in[0];
    const float* W1 = (const float*)d_in[1];
    const float* b1 = (const float*)d_in[2];
    const float* W2 = (const float*)d_in[3];
    const float* b2 = (const float*)d_in[4];
    float* out = (float*)d_out;

    const int B = in_sizes[0] / 4;     // 16384
    const int grid = B / 16;           // 1024 batch tiles
    dqsn_fused_kernel<<<grid, 256, 0, stream>>>(x, W1, b1, W2, b2, out);
}